// IncConvModule_7387343749302
// MI455X (gfx1250) — compile-verified
//
#include <hip/hip_runtime.h>

// ---- problem constants (match setup_inputs) ----
#define BATCH   16
#define CIN     64
#define COUT    64
#define H       224
#define W       224
#define KSZ     3
#define PH      64      // p_height
#define PW      64      // p_width
#define BLK     16      // 16x16 spatial block per workgroup

// LDS tile: 18x18 spatial x 32 input channels, layout [s][ci] with ci-stride
// padded to 36 floats. B fragments (ci, ci+1) are contiguous -> ds_load_b64,
// and 36 ≡ 4 (mod 64, gcd 4) makes the 32-lane b64 read pattern cover all 64
// banks exactly once (conflict-free).
#define CI_CHUNK   32
#define TILE_W     18
#define NSPAT      (TILE_W * TILE_W)    // 324
#define CI_STRIDE  36

typedef float v2f __attribute__((ext_vector_type(2)));
typedef float v8f __attribute__((ext_vector_type(8)));

// Copy the (zero) out_tensor baseline into d_out (faithful to
// dynamic_update_slice semantics outside the patch). float4 streaming.
__global__ void baseline_copy_kernel(const float4* __restrict__ src,
                                     float4* __restrict__ dst, int n4) {
    int i = blockIdx.x * blockDim.x + threadIdx.x;
    int stride = gridDim.x * blockDim.x;
    for (; i < n4; i += stride) dst[i] = src[i];
}

// Implicit-GEMM 3x3 conv patch kernel, fp32 WMMA 16x16x4.
// Grid: BATCH * (PH/16) * (PW/16) blocks, 256 threads (8 waves).
__global__ void __launch_bounds__(256)
inc_conv_patch_kernel(const float* __restrict__ in,      // [B][CIN][H][W]
                      const float* __restrict__ wgt,     // [COUT][CIN][3][3]
                      const float* __restrict__ bias,    // [COUT]
                      float* __restrict__ out,           // [B][COUT][H][W]
                      const int* __restrict__ p_row_start,
                      const int* __restrict__ p_col_start,
                      const int* __restrict__ p_pad) {
    __shared__ float lds[NSPAT * CI_STRIDE];             // 46656 B

    const int prow = *p_row_start;
    const int pcol = *p_col_start;
    const int pad  = *p_pad;

    const int bid = blockIdx.x;
    const int nbx = PW / BLK;                 // 4
    const int nby = PH / BLK;                 // 4
    const int b   = bid / (nbx * nby);
    const int blk = bid % (nbx * nby);
    const int by  = blk / nbx;
    const int bx  = blk % nbx;

    const int tid    = threadIdx.x;
    const int wave   = tid >> 5;
    const int lane   = tid & 31;
    const int hi     = lane >> 4;             // half-wave
    const int lm     = lane & 15;             // = px, = A-row-within-tile, = N index
    const int mtile  = wave & 3;              // which 16 output channels
    const int pybase = (wave >> 2) * 8;       // 8 output rows per wave

    const int co_a   = mtile * 16 + lm;       // A-matrix row for this lane
    const float* wrow = wgt + co_a * (CIN * 9);

    // global origin of the 18x18 input tile for this block
    const int gy0 = prow - pad + by * BLK;
    const int gx0 = pcol - pad + bx * BLK;

    v8f acc[8] = {};

    for (int cic = 0; cic < CIN / CI_CHUNK; ++cic) {
        if (cic) __syncthreads();             // prior reads done before reuse
        // ---- stage input chunk [cic*32 .. +32) into LDS (s fast: coalesced) ----
        const int nelem = CI_CHUNK * NSPAT;               // 10368
        for (int idx = tid; idx < nelem; idx += 256) {
            int ci = idx / NSPAT;
            int s  = idx - ci * NSPAT;
            int iy = s / TILE_W;
            int ix = s - iy * TILE_W;
            int gy = gy0 + iy;
            int gx = gx0 + ix;
            float v = 0.0f;
            if (gy >= 0 && gy < H && gx >= 0 && gx < W)
                v = in[(((long)b * CIN + cic * CI_CHUNK + ci) * H + gy) * W + gx];
            lds[s * CI_STRIDE + ci] = v;
        }
        __syncthreads();

        // ---- 9 shifted GEMMs over this 32-channel K chunk ----
        for (int ky = 0; ky < KSZ; ++ky) {
            for (int kx = 0; kx < KSZ; ++kx) {
                // lane's LDS spatial index for t=0 output row
                const int sbase = (pybase + ky) * TILE_W + (lm + kx);
                for (int kb = 0; kb < CI_CHUNK; kb += 4) {
                    const int ci_l = kb + 2 * hi;              // chunk-local ci
                    const int ci_g = cic * CI_CHUNK + ci_l;    // global ci
                    // A fragment: W[co_a][ci_g..ci_g+1][ky][kx]
                    const float* wp = wrow + ci_g * 9 + ky * 3 + kx;
                    v2f A;
                    A.x = wp[0];
                    A.y = wp[9];
                    // B fragment base: contiguous (ci_l, ci_l+1) -> b64
                    int l0 = sbase * CI_STRIDE + ci_l;
#pragma unroll
                    for (int t = 0; t < 8; ++t) {
                        v2f Bv = *(const v2f*)(lds + l0);
                        acc[t] = __builtin_amdgcn_wmma_f32_16x16x4_f32(
                            /*neg_a=*/false, A, /*neg_b=*/false, Bv,
                            /*c_mod=*/(short)0, acc[t],
                            /*reuse_a=*/false, /*reuse_b=*/false);
                        l0 += TILE_W * CI_STRIDE;              // next output row
                    }
                }
            }
        }
    }

    // ---- epilogue: bias + store patch ----
    const int gx = pcol + bx * BLK + lm;
#pragma unroll
    for (int t = 0; t < 8; ++t) {
        const int gy = prow + by * BLK + pybase + t;
#pragma unroll
        for (int r = 0; r < 8; ++r) {
            const int co = mtile * 16 + r + 8 * hi;    // C/D layout: M = r + 8*hi
            out[(((long)b * COUT + co) * H + gy) * W + gx] = acc[t][r] + bias[co];
        }
    }
}

extern "C" void kernel_launch(void* const* d_in, const int* in_sizes, int n_in,
                              void* d_out, int out_size, void* d_ws, size_t ws_size,
                              hipStream_t stream) {
    const float* in_tensor  = (const float*)d_in[0];
    const float* weights    = (const float*)d_in[1];
    const float* biases     = (const float*)d_in[2];
    const float* out_base   = (const float*)d_in[3];
    const int*   p_row      = (const int*)d_in[7];
    const int*   p_col      = (const int*)d_in[8];
    const int*   p_pad      = (const int*)d_in[4];
    float*       out        = (float*)d_out;

    // 1) baseline: out = out_tensor (zeros) everywhere
    int n4 = out_size / 4;
    baseline_copy_kernel<<<8192, 256, 0, stream>>>(
        (const float4*)out_base, (float4*)out, n4);

    // 2) compute 64x64 patch with WMMA implicit GEMM (overwrites patch region)
    int nblocks = BATCH * (PH / BLK) * (PW / BLK);   // 256
    inc_conv_patch_kernel<<<nblocks, 256, 0, stream>>>(
        in_tensor, weights, biases, out, p_row, p_col, p_pad);
}